// ChamferDist_28870770163805
// MI455X (gfx1250) — compile-verified
//
#include <hip/hip_runtime.h>
#include <hip/hip_bf16.h>

typedef __attribute__((ext_vector_type(2))) float v2f;
typedef __attribute__((ext_vector_type(8))) float v8f;

#define BSZ   64
#define FDIM  4
#define NPTS  2048
#define NTILE (NPTS / 16)   // 128 column tiles
#define BIGV  1e30f

// One wave (32 lanes) owns one (batch, rowTile) pair: a 16-row strip of the
// NxN distance matrix. It loops over all 128 column tiles; each tile is one
// V_WMMA_F32_16X16X4_F32. A is pre-scaled by -2 so D = -2 * X.Y^T directly.
// Masked columns contribute BIG so they never win the min; row masking is
// applied later in the reduction (min * mask), matching the reference.
__global__ __launch_bounds__(256) void chamfer_rowmin_kernel(
    const float* __restrict__ X,     // [B, F, N] "rows" of d2
    const float* __restrict__ Y,     // [B, F, N] "cols" of d2
    const int* __restrict__ mask,    // [B, N] gates columns
    float* __restrict__ outMin)      // [B, N] per-row min over valid cols
{
    const int lane   = threadIdx.x & 31;
    const int waveIb = threadIdx.x >> 5;
    const int wid    = blockIdx.x * 8 + waveIb;   // 8 waves / block
    const int b      = wid >> 7;                  // wid / NTILE
    const int rowT   = wid & (NTILE - 1);
    const int half   = lane >> 4;                 // 0: lanes 0-15, 1: lanes 16-31
    const int l16    = lane & 15;

    const float* Xb = X + (size_t)b * FDIM * NPTS;
    const float* Yb = Y + (size_t)b * FDIM * NPTS;
    const int*   mb = mask + (size_t)b * NPTS;

    // ---- A tile: 16 rows x 4 (f32). Lane layout per ISA: lanes 0-15 hold
    // K=0,1 in VGPR0,1; lanes 16-31 hold K=2,3. Pre-scale by -2.
    const int row = rowT * 16 + l16;
    const float x0 = Xb[0 * NPTS + row];
    const float x1 = Xb[1 * NPTS + row];
    const float x2 = Xb[2 * NPTS + row];
    const float x3 = Xb[3 * NPTS + row];
    const float xn = x0 * x0 + x1 * x1 + x2 * x2 + x3 * x3;

    v2f a;
    a.x = -2.0f * (half ? x2 : x0);
    a.y = -2.0f * (half ? x3 : x1);

    // Broadcast |x_row|^2 to match the D accumulator row mapping:
    // D VGPR i, lanes 0-15 -> row i; lanes 16-31 -> row i+8.
    float xrow[8];
#pragma unroll
    for (int i = 0; i < 8; ++i)
        xrow[i] = __shfl(xn, half ? (i + 8) : i, 16);

    float rm[8];
#pragma unroll
    for (int i = 0; i < 8; ++i) rm[i] = BIGV;

    for (int mt = 0; mt < NTILE; ++mt) {
        const int col = mt * 16 + l16;
        const float y0 = Yb[0 * NPTS + col];
        const float y1 = Yb[1 * NPTS + col];
        const float y2 = Yb[2 * NPTS + col];
        const float y3 = Yb[3 * NPTS + col];
        float yn = y0 * y0 + y1 * y1 + y2 * y2 + y3 * y3;
        yn = mb[col] ? yn : BIGV;   // masked column -> never wins the min

        v2f bb;
        bb.x = half ? y2 : y0;
        bb.y = half ? y3 : y1;

        v8f c = {0.f, 0.f, 0.f, 0.f, 0.f, 0.f, 0.f, 0.f};
        // D = (-2X) * Y^T   (8 args: neg_a, A, neg_b, B, c_mod, C, reuse_a, reuse_b)
        v8f d = __builtin_amdgcn_wmma_f32_16x16x4_f32(
            false, a, false, bb, (short)0, c, false, false);

#pragma unroll
        for (int i = 0; i < 8; ++i) {
            float v = d[i] + xrow[i] + yn;   // |x|^2 + |y|^2 - 2 x.y
            v = fmaxf(v, 0.0f);
            rm[i] = fminf(rm[i], v);
        }
    }

    // Cross-lane min within each 16-lane half: all 16 lanes of a half hold
    // partial mins (over cols == l16 mod 16) for the same 8 rows.
#pragma unroll
    for (int i = 0; i < 8; ++i) {
        float v = rm[i];
        v = fminf(v, __shfl_xor(v, 1, 16));
        v = fminf(v, __shfl_xor(v, 2, 16));
        v = fminf(v, __shfl_xor(v, 4, 16));
        v = fminf(v, __shfl_xor(v, 8, 16));
        rm[i] = v;
    }

    if (l16 == 0) {
        float* o = outMin + (size_t)b * NPTS + rowT * 16 + half * 8;
#pragma unroll
        for (int i = 0; i < 8; ++i) o[i] = rm[i];
    }
}

// Per-batch masked sums -> loss_per[b] = (sum(minA*m) + sum(minB*m)) / cnt
__global__ __launch_bounds__(256) void chamfer_reduce_kernel(
    const float* __restrict__ minA, const float* __restrict__ minB,
    const int* __restrict__ mask, float* __restrict__ lossPer)
{
    const int b   = blockIdx.x;
    const int tid = threadIdx.x;
    __shared__ float sA[256], sB[256], sC[256];

    float a = 0.f, bb = 0.f, c = 0.f;
    for (int n = tid; n < NPTS; n += 256) {
        const float mf = (float)mask[(size_t)b * NPTS + n];
        a  += minA[(size_t)b * NPTS + n] * mf;
        bb += minB[(size_t)b * NPTS + n] * mf;
        c  += mf;
    }
    sA[tid] = a; sB[tid] = bb; sC[tid] = c;
    __syncthreads();
    for (int s = 128; s > 0; s >>= 1) {
        if (tid < s) {
            sA[tid] += sA[tid + s];
            sB[tid] += sB[tid + s];
            sC[tid] += sC[tid + s];
        }
        __syncthreads();
    }
    if (tid == 0) lossPer[b] = sA[0] / sC[0] + sB[0] / sC[0];
}

__global__ void chamfer_final_kernel(const float* __restrict__ lossPer,
                                     float* __restrict__ out)
{
    if (threadIdx.x == 0) {
        float s = 0.f;
        for (int b = 0; b < BSZ; ++b) s += lossPer[b];
        out[0] = s / (float)BSZ;
    }
}

extern "C" void kernel_launch(void* const* d_in, const int* in_sizes, int n_in,
                              void* d_out, int out_size, void* d_ws, size_t ws_size,
                              hipStream_t stream) {
    const float* recon  = (const float*)d_in[0];  // [B, F, N] f32
    const float* target = (const float*)d_in[1];  // [B, F, N] f32
    const int*   mask   = (const int*)d_in[2];    // [B, N] i32
    float* out = (float*)d_out;

    float* minA    = (float*)d_ws;                // B*N
    float* minB    = minA + (size_t)BSZ * NPTS;   // B*N
    float* lossPer = minB + (size_t)BSZ * NPTS;   // B

    const int nBlocks = (BSZ * NTILE) / 8;        // 8 waves per 256-thread block

    // Pass 1: min over valid target points for each recon row
    chamfer_rowmin_kernel<<<nBlocks, 256, 0, stream>>>(recon, target, mask, minA);
    // Pass 2: min over valid recon points for each target row (transposed role)
    chamfer_rowmin_kernel<<<nBlocks, 256, 0, stream>>>(target, recon, mask, minB);
    // Per-batch masked reduction, then mean
    chamfer_reduce_kernel<<<BSZ, 256, 0, stream>>>(minA, minB, mask, lossPer);
    chamfer_final_kernel<<<1, 64, 0, stream>>>(lossPer, out);
}